// VGGLoss_36189394436760
// MI455X (gfx1250) — compile-verified
//
#include <hip/hip_runtime.h>
#include <hip/hip_bf16.h>
#include <cmath>

typedef __attribute__((ext_vector_type(2)))  float    v2f;
typedef __attribute__((ext_vector_type(4)))  float    v4f;
typedef __attribute__((ext_vector_type(8)))  float    v8f;
typedef __attribute__((ext_vector_type(4)))  int      v4i;

#define N_BATCH 4
#define C_CH    256
#define P_SPAT  4096   // 64*64
#define EPS_K   1e-5f

// ---------------------------------------------------------------------------
// Async global->LDS staging (gfx1250 GLOBAL_LOAD_ASYNC_TO_LDS_B128, ASYNCcnt)
// Builtin signature (from clang diagnostic): (v4i AS1*, v4i AS3*, Imm, Imm).
// ---------------------------------------------------------------------------
#if __has_builtin(__builtin_amdgcn_global_load_async_to_lds_b128)
#define HAVE_ASYNC_LDS 1
typedef __attribute__((address_space(1))) v4i* gv4i_p;
typedef __attribute__((address_space(3))) v4i* lv4i_p;
#else
#define HAVE_ASYNC_LDS 0
#endif

__device__ __forceinline__ void stage_b128(const float* __restrict__ g,
                                           float* __restrict__ lds) {
#if HAVE_ASYNC_LDS
    __builtin_amdgcn_global_load_async_to_lds_b128(
        (gv4i_p)const_cast<float*>(g), (lv4i_p)lds, 0, 0);
#else
    *(v4f*)lds = *(const v4f*)g;
#endif
}

__device__ __forceinline__ void stage_join() {
#if HAVE_ASYNC_LDS
#if __has_builtin(__builtin_amdgcn_s_wait_asynccnt)
    __builtin_amdgcn_s_wait_asynccnt(0);
#else
    asm volatile("s_wait_asynccnt 0x0" ::: "memory");
#endif
#endif
}

// ---------------------------------------------------------------------------
// Kernel 1: channel sums sumX[n,p] = sum_c X[n,c,p] via V_WMMA_F32_16X16X4_F32
// One wave per 16-spatial-row tile; A = 16x4 channel slab (fp32, exact),
// B = all-ones so every D column equals the row sums (B/D layout-proof).
// 2048 tiles: [0,1024) -> T_features, [1024,2048) -> I_features.
// ---------------------------------------------------------------------------
__global__ __launch_bounds__(256) void k_chansum(const float* __restrict__ Tf,
                                                 const float* __restrict__ If,
                                                 float* __restrict__ sumT,
                                                 float* __restrict__ sumI) {
    const int wave = (blockIdx.x * blockDim.x + threadIdx.x) >> 5;
    const int lane = threadIdx.x & 31;
    int tile = wave;
    const float* src;
    float* dst;
    if (tile < 1024) { src = Tf; dst = sumT; }
    else             { tile -= 1024; src = If; dst = sumI; }
    const int n   = tile >> 8;           // 256 tiles per batch image
    const int p0  = (tile & 255) << 4;   // 16 spatial rows per tile
    const int hi  = lane >> 4;           // lane half selects K-pair
    const int row = lane & 15;           // M = row
    const int p   = p0 + row;

    v8f acc = {};
    v2f b = {1.0f, 1.0f};
    const float* ptr = src + (size_t)n * (C_CH * P_SPAT) + (hi * 2) * P_SPAT + p;
    #pragma unroll 4
    for (int kk = 0; kk < 64; ++kk) {
        v2f a;
        a.x = ptr[0];        // K = 4*kk + 2*hi
        a.y = ptr[P_SPAT];   // K = 4*kk + 2*hi + 1
        acc = __builtin_amdgcn_wmma_f32_16x16x4_f32(false, a, false, b,
                                                    (short)0, acc, false, false);
        ptr += 4 * P_SPAT;
    }
    // D: lanes 0-15 vgpr r -> M=r, lanes 16-31 vgpr r -> M=8+r; all columns
    // identical, so lane 0 writes rows 0..7 and lane 16 rows 8..15.
    if (row == 0) {
        float* o = dst + n * P_SPAT + p0 + hi * 8;
        #pragma unroll
        for (int r = 0; r < 8; ++r) o[r] = acc[r];
    }
}

// ---------------------------------------------------------------------------
// Kernel 2: for 64 j's per block, min over i of |sumI[i]-sumT[j]| split into
// 4 i-chunks (chunk uniform per wave -> LDS broadcast reads). Emits
// interleaved (sumT[j], 1/(min+eps)) pairs for kernel 3.
// ---------------------------------------------------------------------------
__global__ __launch_bounds__(256) void k_div(const float* __restrict__ sumI,
                                             const float* __restrict__ sumT,
                                             float* __restrict__ tdpair) {
    __shared__ float sI[P_SPAT];
    __shared__ float red[256];
    const int tid = threadIdx.x;
    const int n   = blockIdx.x >> 6;          // 64 blocks per batch
    const int jb  = (blockIdx.x & 63) << 6;   // 64 j's per block
    for (int t = tid; t < P_SPAT / 4; t += 256)
        stage_b128(sumI + n * P_SPAT + 4 * t, sI + 4 * t);
    stage_join();
    __syncthreads();

    const int jl    = tid & 63;
    const int chunk = tid >> 6;               // uniform within a wave
    const float tj  = sumT[n * P_SPAT + jb + jl];
    float m = 3.4e38f;
    const int i0 = chunk * (P_SPAT / 4);
    #pragma unroll 8
    for (int i = i0; i < i0 + P_SPAT / 4; ++i)
        m = fminf(m, fabsf(sI[i] - tj));
    red[tid] = m;
    __syncthreads();
    if (tid < 64) {
        float mm = fminf(fminf(red[tid], red[tid + 64]),
                         fminf(red[tid + 128], red[tid + 192]));
        float* o = tdpair + 2 * (n * P_SPAT + jb + tid);
        o[0] = sumT[n * P_SPAT + jb + tid];
        o[1] = 1.0f / (mm + EPS_K);
    }
}

// ---------------------------------------------------------------------------
// Kernel 3: per i: m = min_j |sumI[i]-sumT[j]| * dscale[j] over interleaved
// (t,d) pairs (one ds_load_b64 broadcast per j), 4 j-chunks per i; then
// exp(1-m) summed into acc[n] via global fp32 atomics.
// ---------------------------------------------------------------------------
__global__ __launch_bounds__(256) void k_kmax(const float* __restrict__ sumI,
                                              const float* __restrict__ tdpair,
                                              float* __restrict__ acc) {
    __shared__ float sTD[2 * P_SPAT];
    __shared__ float red[256];
    const int tid = threadIdx.x;
    const int n   = blockIdx.x >> 6;
    const int ib  = (blockIdx.x & 63) << 6;
    for (int t = tid; t < (2 * P_SPAT) / 4; t += 256)
        stage_b128(tdpair + 2 * n * P_SPAT + 4 * t, sTD + 4 * t);
    stage_join();
    __syncthreads();

    const int il    = tid & 63;
    const int chunk = tid >> 6;               // uniform within a wave
    const float si  = sumI[n * P_SPAT + ib + il];
    float m = 3.4e38f;
    const int j0 = chunk * (P_SPAT / 4);
    #pragma unroll 8
    for (int j = j0; j < j0 + P_SPAT / 4; ++j) {
        float t = sTD[2 * j];
        float d = sTD[2 * j + 1];
        m = fminf(m, fabsf(si - t) * d);
    }
    red[tid] = m;
    __syncthreads();
    float val = 0.0f;
    if (tid < 64) {
        float mm = fminf(fminf(red[tid], red[tid + 64]),
                         fminf(red[tid + 128], red[tid + 192]));
        val = expf(1.0f - mm);
    }
    __syncthreads();
    red[tid] = val;                           // threads >= 64 contribute 0
    __syncthreads();
    for (int s = 128; s > 0; s >>= 1) {
        if (tid < s) red[tid] += red[tid + s];
        __syncthreads();
    }
    if (tid == 0) atomicAdd(&acc[n], red[0]);
}

// ---------------------------------------------------------------------------
// Kernel 4: scalar epilogue: mean_n( -log( acc[n] / P ) )
// ---------------------------------------------------------------------------
__global__ void k_final(const float* __restrict__ acc, float* __restrict__ out) {
    if (threadIdx.x == 0) {
        float s = 0.0f;
        for (int n = 0; n < N_BATCH; ++n)
            s += -logf(acc[n] / (float)P_SPAT);
        out[0] = s / (float)N_BATCH;
    }
}

extern "C" void kernel_launch(void* const* d_in, const int* in_sizes, int n_in,
                              void* d_out, int out_size, void* d_ws, size_t ws_size,
                              hipStream_t stream) {
    const float* Tf = (const float*)d_in[0];  // T_features [4,256,64,64] fp32
    const float* If = (const float*)d_in[1];  // I_features [4,256,64,64] fp32
    float* ws     = (float*)d_ws;
    float* sumT   = ws;               // 16384 floats
    float* sumI   = ws + 16384;       // 16384 floats
    float* tdpair = ws + 32768;       // 32768 floats (interleaved t,d pairs)
    float* acc    = ws + 65536;       // 4 floats (zeroed every call)

    (void)hipMemsetAsync(acc, 0, N_BATCH * sizeof(float), stream);
    k_chansum<<<256, 256, 0, stream>>>(Tf, If, sumT, sumI);
    k_div    <<<256, 256, 0, stream>>>(sumI, sumT, tdpair);
    k_kmax   <<<256, 256, 0, stream>>>(sumI, tdpair, acc);
    k_final  <<<1,   32,  0, stream>>>(acc, (float*)d_out);
}